// My_impute_Loss_27161373180458
// MI455X (gfx1250) — compile-verified
//
#include <hip/hip_runtime.h>
#include <hip/hip_bf16.h>

// Problem constants (from reference): preds/targets are [N, C] f32, row-major.
#define NROWS 16384
#define NCOLS 4096

typedef __attribute__((ext_vector_type(2))) float v2f;
typedef __attribute__((ext_vector_type(4))) float v4f;
typedef __attribute__((ext_vector_type(8))) float v8f;

// ---------------------------------------------------------------------------
// Phase 1: streaming pass. Each thread owns 4 consecutive columns, loops over
// this block's row chunk, accumulating 5 moments. Fully coalesced 16B
// non-temporal loads (data is streamed exactly once; NT keeps L2 clean for
// the partials). Partials written which-major: part[w*R*C + chunk*C + col].
// ---------------------------------------------------------------------------
__global__ __launch_bounds__(128) void colstats_partial(
    const float* __restrict__ preds, const float* __restrict__ targets,
    float* __restrict__ part, int rowsPerChunk, unsigned long long whichStride) {
  const int col = blockIdx.x * 512 + threadIdx.x * 4;
  const int r0 = blockIdx.y * rowsPerChunk;

  v4f sp = 0.0f, st = 0.0f, spp = 0.0f, stt = 0.0f, spt = 0.0f;

  const float* __restrict__ pbase = preds + (size_t)r0 * NCOLS + col;
  const float* __restrict__ tbase = targets + (size_t)r0 * NCOLS + col;

#pragma unroll 4
  for (int r = 0; r < rowsPerChunk; ++r) {
    v4f p = __builtin_nontemporal_load(
        reinterpret_cast<const v4f*>(pbase + (size_t)r * NCOLS));
    v4f t = __builtin_nontemporal_load(
        reinterpret_cast<const v4f*>(tbase + (size_t)r * NCOLS));
    sp += p;
    st += t;
    spp += p * p;
    stt += t * t;
    spt += p * t;
  }

  float* dst = part + (size_t)blockIdx.y * NCOLS + col;
  *reinterpret_cast<v4f*>(dst + 0ull * whichStride) = sp;
  *reinterpret_cast<v4f*>(dst + 1ull * whichStride) = st;
  *reinterpret_cast<v4f*>(dst + 2ull * whichStride) = spp;
  *reinterpret_cast<v4f*>(dst + 3ull * whichStride) = stt;
  *reinterpret_cast<v4f*>(dst + 4ull * whichStride) = spt;
}

// ---------------------------------------------------------------------------
// Phase 2: reduce the R partial chunks per column with the matrix unit:
//   colsums(1xC) = ones(1xR) x part(RxC)
// done as accumulating V_WMMA_F32_16X16X4_F32 chains, A = ones(16x4).
// D[m][n] = sum_k B[k][n] + C[m][n]  -> every row of D equals the column sums.
// B layout (f32 4x16): N = lane%16; lane-halves/regs carry the 4 K values
// (with A == 1 the sum is invariant to the exact K permutation).
// Then per-lane: compute PCC + cosine for column (colb + lane%16); every
// column appears in exactly 2 lanes, so weight 0.5 (branchless, EXEC stays
// all-ones as WMMA requires). LDS tree-reduce -> blockSums[blockIdx].
// ---------------------------------------------------------------------------
__global__ __launch_bounds__(128) void colstats_finalize(
    const float* __restrict__ part, float* __restrict__ blockSums, int R,
    unsigned long long whichStride) {
  const int tid = threadIdx.x;
  const int lane = tid & 31;
  const int wave = tid >> 5;
  const int n = lane & 15;
  const int koff = (lane < 16) ? 0 : 2;

  v2f a;  // A = ones(16x4): 2 VGPRs, all 1.0
  a.x = 1.0f;
  a.y = 1.0f;
  const v8f vzero = {0.f, 0.f, 0.f, 0.f, 0.f, 0.f, 0.f, 0.f};

  float lossAcc = 0.0f;

  for (int strip = wave; strip < 16; strip += 4) {
    const int colb = blockIdx.x * 256 + strip * 16;

    v8f acc0 = vzero, acc1 = vzero, acc2 = vzero, acc3 = vzero, acc4 = vzero;

    for (int kk = 0; kk < R; kk += 4) {
      const int k0 = kk + koff;
      const float* base = part + (size_t)k0 * NCOLS + colb + n;
      v2f b0, b1, b2, b3, b4;
      b0.x = base[0ull * whichStride];
      b0.y = base[0ull * whichStride + NCOLS];
      b1.x = base[1ull * whichStride];
      b1.y = base[1ull * whichStride + NCOLS];
      b2.x = base[2ull * whichStride];
      b2.y = base[2ull * whichStride + NCOLS];
      b3.x = base[3ull * whichStride];
      b3.y = base[3ull * whichStride + NCOLS];
      b4.x = base[4ull * whichStride];
      b4.y = base[4ull * whichStride + NCOLS];
      acc0 = __builtin_amdgcn_wmma_f32_16x16x4_f32(false, a, false, b0,
                                                   (short)0, acc0, false, false);
      acc1 = __builtin_amdgcn_wmma_f32_16x16x4_f32(false, a, false, b1,
                                                   (short)0, acc1, false, false);
      acc2 = __builtin_amdgcn_wmma_f32_16x16x4_f32(false, a, false, b2,
                                                   (short)0, acc2, false, false);
      acc3 = __builtin_amdgcn_wmma_f32_16x16x4_f32(false, a, false, b3,
                                                   (short)0, acc3, false, false);
      acc4 = __builtin_amdgcn_wmma_f32_16x16x4_f32(false, a, false, b4,
                                                   (short)0, acc4, false, false);
    }

    // Every lane's acc*[0] holds the full column sum for column (colb + n).
    const float Sp = acc0[0];
    const float St = acc1[0];
    const float Spp = acc2[0];
    const float Stt = acc3[0];
    const float Spt = acc4[0];

    const float invN = 1.0f / (float)NROWS;
    const float mup = Sp * invN;
    const float mut = St * invN;
    const float cov = Spt * invN - mup * mut;
    const float varp = fmaxf(Spp * invN - mup * mup, 0.0f);
    const float vart = fmaxf(Stt * invN - mut * mut, 0.0f);
    const float denom = sqrtf(varp) * sqrtf(vart);
    const float pcc = (denom > 0.0f) ? (cov / denom) : 0.0f;  // NaN -> 0

    const float np = fmaxf(sqrtf(Spp), 1e-8f);
    const float nt = fmaxf(sqrtf(Stt), 1e-8f);
    const float cosv = Spt / (np * nt);

    lossAcc += 0.5f * ((1.0f - pcc) + (1.0f - cosv));
  }

  __shared__ float red[128];
  red[tid] = lossAcc;
  __syncthreads();
#pragma unroll
  for (int s = 64; s > 0; s >>= 1) {
    if (tid < s) red[tid] += red[tid + s];
    __syncthreads();
  }
  if (tid == 0) blockSums[blockIdx.x] = red[0];
}

// ---------------------------------------------------------------------------
// Phase 3: combine 16 block sums -> scalar mean over 4096 columns.
// ---------------------------------------------------------------------------
__global__ void colstats_write(const float* __restrict__ blockSums,
                               float* __restrict__ out) {
  if (threadIdx.x == 0) {
    float s = 0.0f;
#pragma unroll
    for (int i = 0; i < 16; ++i) s += blockSums[i];
    out[0] = s * (1.0f / (float)NCOLS);
  }
}

extern "C" void kernel_launch(void* const* d_in, const int* in_sizes, int n_in,
                              void* d_out, int out_size, void* d_ws,
                              size_t ws_size, hipStream_t stream) {
  const float* preds = (const float*)d_in[0];
  const float* targets = (const float*)d_in[1];
  float* ws = (float*)d_ws;
  float* out = (float*)d_out;

  // Pick R (row chunks) to fit partials (5*R*C floats + 16 block sums) in ws.
  // Deterministic for a fixed ws_size. R must be a multiple of 4 (WMMA K-tile)
  // and divide NROWS; all candidates are powers of two.
  int R = 4;
  const int candidates[5] = {64, 32, 16, 8, 4};
  for (int i = 0; i < 5; ++i) {
    size_t need = (size_t)5 * candidates[i] * NCOLS * sizeof(float) +
                  16 * sizeof(float);
    if (ws_size >= need) {
      R = candidates[i];
      break;
    }
  }
  const unsigned long long whichStride = (unsigned long long)R * NCOLS;
  float* part = ws;
  float* blockSums = ws + (size_t)5 * whichStride;

  dim3 g1(NCOLS / 512, R);  // 8 column tiles x R row chunks
  colstats_partial<<<g1, 128, 0, stream>>>(preds, targets, part, NROWS / R,
                                           whichStride);
  colstats_finalize<<<16, 128, 0, stream>>>(part, blockSums, R, whichStride);
  colstats_write<<<1, 32, 0, stream>>>(blockSums, out);
}